// RelationGraphConvolution_noBasisRegularization_16982300688782
// MI455X (gfx1250) — compile-verified
//
#include <hip/hip_runtime.h>

typedef __attribute__((ext_vector_type(2))) float v2f;
typedef __attribute__((ext_vector_type(8))) float v8f;

#define N_NODES 50000
#define D_IN    512
#define D_OUT   256
#define N_REL   8
#define N_EDGE  500000

// ---------------------------------------------------------------------------
// GEMM: pre = x @ Wr      x:[N,512] row-major, Wr:[512,256] row-major.
// Block: 256 thr = 8 waves, block tile 128x256, wave tile 64x64.
// Wave-level fragments per CDNA5 ISA 7.12.2 (wave32):
//   A (16x4 f32, 2 VGPR):  lane(half,lr): v_j = A[M=lr][K = 2*half + j]
//   B (4x16 f32, 2 VGPR):  lane(half,lr): v_j = B[K = 2*half + j][N=lr]
//   C/D (16x16 f32, 8 VGPR): v_j = C[M = j + 8*half][N=lr]
// ---------------------------------------------------------------------------
__global__ __launch_bounds__(256)
void rgcn_gemm_f32_wmma(const float* __restrict__ x,
                        const float* __restrict__ Wr,
                        float* __restrict__ pre) {
  const int lane = threadIdx.x & 31;
  const int wave = threadIdx.x >> 5;
  const int half = lane >> 4;     // 0..1
  const int lr   = lane & 15;     // 0..15
  const int wm   = wave >> 2;     // 0..1 (rows)
  const int wn   = wave & 3;      // 0..3 (cols)
  const int row0 = blockIdx.x * 128 + wm * 64;
  const int col0 = wn * 64;

  v8f acc[4][4];
#pragma unroll
  for (int mi = 0; mi < 4; ++mi)
#pragma unroll
    for (int ni = 0; ni < 4; ++ni)
      acc[mi][ni] = v8f{};

  // Per-fragment base pointers (k-invariant part of the address).
  const float* pa[4];
#pragma unroll
  for (int mi = 0; mi < 4; ++mi) {
    int row = row0 + 16 * mi + lr;
    row = row < N_NODES ? row : (N_NODES - 1);   // clamp OOB loads
    pa[mi] = x + (size_t)row * D_IN + 2 * half;
  }
  const float* pb[4];
#pragma unroll
  for (int ni = 0; ni < 4; ++ni) {
    const int col = col0 + 16 * ni + lr;
    pb[ni] = Wr + (size_t)(2 * half) * D_OUT + col;
  }

  for (int k = 0; k < D_IN; k += 4) {
    v2f a[4], b[4];
#pragma unroll
    for (int mi = 0; mi < 4; ++mi)
      a[mi] = *(const v2f*)(pa[mi] + k);          // x[row][k+2h], x[row][k+2h+1]
#pragma unroll
    for (int ni = 0; ni < 4; ++ni) {
      const float* p = pb[ni] + (size_t)k * D_OUT;
      b[ni].x = p[0];                              // Wr[k+2h  ][col]
      b[ni].y = p[D_OUT];                          // Wr[k+2h+1][col]
    }
#pragma unroll
    for (int mi = 0; mi < 4; ++mi)
#pragma unroll
      for (int ni = 0; ni < 4; ++ni)
        acc[mi][ni] = __builtin_amdgcn_wmma_f32_16x16x4_f32(
            /*neg_a=*/false, a[mi], /*neg_b=*/false, b[ni],
            /*c_mod=*/(short)0, acc[mi][ni],
            /*reuse_a=*/false, /*reuse_b=*/false);
  }

  // Store D tiles (coalesced across lanes along N).
#pragma unroll
  for (int mi = 0; mi < 4; ++mi) {
#pragma unroll
    for (int v = 0; v < 8; ++v) {
      const int row = row0 + 16 * mi + 8 * half + v;
      if (row < N_NODES) {
#pragma unroll
        for (int ni = 0; ni < 4; ++ni) {
          const int col = col0 + 16 * ni + lr;
          pre[(size_t)row * D_OUT + col] = acc[mi][ni][v];
        }
      }
    }
  }
}

// ---------------------------------------------------------------------------
// Scatter: out[dst] += w * pre[src]   (one edge per block-iteration,
// 256 threads cover the 256-wide feature row; edge metadata is uniform ->
// scalar loads; fp32 atomic add, relaxed/agent, no return value).
// pre (51.2 MB) and out (51.2 MB) both stay resident in the 192 MB L2.
// ---------------------------------------------------------------------------
__global__ __launch_bounds__(256)
void rgcn_scatter(const float* __restrict__ pre,
                  const int* __restrict__ esrc,
                  const int* __restrict__ edst,
                  const float* __restrict__ ew,
                  float* __restrict__ out) {
  const int t = threadIdx.x;
  for (int e = blockIdx.x; e < N_EDGE; e += gridDim.x) {
    const int   s = esrc[e];
    const int   d = edst[e];
    const float w = ew[e];
    const float v = pre[(size_t)s * D_OUT + t] * w;
    __hip_atomic_fetch_add(&out[(size_t)d * D_OUT + t], v,
                           __ATOMIC_RELAXED, __HIP_MEMORY_SCOPE_AGENT);
  }
}

__global__ void rgcn_zero(float* __restrict__ p, int n) {
  const int i = blockIdx.x * blockDim.x + threadIdx.x;
  if (i < n) p[i] = 0.0f;
}

__global__ void rgcn_relu(float* __restrict__ p, int n) {
  const int i = blockIdx.x * blockDim.x + threadIdx.x;
  if (i < n) p[i] = fmaxf(p[i], 0.0f);
}

// ---------------------------------------------------------------------------
// Inputs (setup_inputs order):
//   0: x           [N, D_IN]      f32
//   1: W           [R, D_IN, D_OUT] f32
//   2: edge_weight [R, E]         f32
//   3: edge_src    [R, E]         i32
//   4: edge_dst    [R, E]         i32
// Output: [N, D_OUT] f32.  Workspace: one relation's pre_sup (51.2 MB).
// ---------------------------------------------------------------------------
extern "C" void kernel_launch(void* const* d_in, const int* in_sizes, int n_in,
                              void* d_out, int out_size, void* d_ws, size_t ws_size,
                              hipStream_t stream) {
  const float* x    = (const float*)d_in[0];
  const float* W    = (const float*)d_in[1];
  const float* ew   = (const float*)d_in[2];
  const int*   esrc = (const int*)d_in[3];
  const int*   edst = (const int*)d_in[4];
  float* out = (float*)d_out;
  float* pre = (float*)d_ws;                 // N_NODES * D_OUT floats

  const int outN = N_NODES * D_OUT;
  rgcn_zero<<<(outN + 255) / 256, 256, 0, stream>>>(out, outN);

  const dim3 gemmGrid((N_NODES + 127) / 128);   // 391 blocks of 8 waves
  for (int r = 0; r < N_REL; ++r) {
    // Alternate GEMM -> scatter so pre_sup is L2-hot when gathered;
    // stream ordering provides the gemm->scatter->next-gemm dependency.
    rgcn_gemm_f32_wmma<<<gemmGrid, 256, 0, stream>>>(
        x, W + (size_t)r * D_IN * D_OUT, pre);
    rgcn_scatter<<<16384, 256, 0, stream>>>(
        pre, esrc + (size_t)r * N_EDGE, edst + (size_t)r * N_EDGE,
        ew + (size_t)r * N_EDGE, out);
  }

  rgcn_relu<<<(outN + 255) / 256, 256, 0, stream>>>(out, outN);
}